// DeFeat_64965675320040
// MI455X (gfx1250) — compile-verified
//
#include <hip/hip_runtime.h>
#include <hip/hip_bf16.h>
#include <math.h>

typedef __attribute__((ext_vector_type(2))) float v2f;
typedef __attribute__((ext_vector_type(8))) float v8f;

#define CCH 256          // channels
#define BATCH 8
#define NBOX 16
#define WGT_GT 0.004f
#define WGT_BG 0.0002f

// ---------------------------------------------------------------------------
// zero the 10 per-level accumulators (5 levels x {gt, bg})
// ---------------------------------------------------------------------------
__global__ void defeat_init_acc(float* acc) {
    if (threadIdx.x < 10) acc[threadIdx.x] = 0.0f;
}

// ---------------------------------------------------------------------------
// Per-level kernel (templated on H and STRIDE, both powers of two so all
// index math folds to shifts/masks). One wave32 computes a 16(out-ch) x
// 64(pixel) tile of the 1x1-conv GEMM with V_WMMA_F32_16X16X4_F32, then the
// masked squared diff against the teacher, then atomically accumulates
// s_gt / s_bg for its level.
// ---------------------------------------------------------------------------
template <int H, int STRIDE>
__global__ __launch_bounds__(256) void defeat_level_kernel(
    const float* __restrict__ S,      // student  [B, C, H, H]
    const float* __restrict__ T,      // teacher  [B, C, H, H]
    const float* __restrict__ Wm,     // adapt weight [C, C] (o, c) row-major
    const float* __restrict__ bias,   // adapt bias  [C]
    const float* __restrict__ boxes,  // gt boxes [B, 16, 4]
    float* __restrict__ acc,          // accumulators, 2 per level
    int totalWaves, int level)
{
    constexpr int HW      = H * H;
    constexpr int NTILES  = HW >> 6;          // 64-pixel tiles (power of two)
    constexpr int PERB    = 16 * NTILES;      // waves per batch image

    const int lane = threadIdx.x & 31;
    const int wave = (blockIdx.x * blockDim.x + threadIdx.x) >> 5;
    if (wave >= totalWaves) return;           // uniform per-wave: EXEC stays all-ones

    const int b     = wave / PERB;            // PERB is constexpr pow2 -> shift
    const int rem   = wave & (PERB - 1);
    const int ntile = rem >> 4;               // mtile innermost -> S-tile L2 reuse
    const int mtile = rem & 15;

    const int l16 = lane & 15;                // N / M position within fragment
    const int hi  = lane >> 4;                // K sub-pair selector (0 or 1)

    const float* __restrict__ Sb = S + (size_t)b * CCH * HW;
    const float* __restrict__ Tb = T + (size_t)b * CCH * HW;

    // A fragment source: row of W, contiguous float2 per lane
    const float* __restrict__ Arow = Wm + (size_t)(mtile * 16 + l16) * CCH;

    // B fragment sources: one column of S per subtile
    const int p0 = ntile * 64;
    const float* __restrict__ Bcol0 = Sb + p0 + 0 * 16 + l16;
    const float* __restrict__ Bcol1 = Sb + p0 + 1 * 16 + l16;
    const float* __restrict__ Bcol2 = Sb + p0 + 2 * 16 + l16;
    const float* __restrict__ Bcol3 = Sb + p0 + 3 * 16 + l16;

    v8f acc0 = {}; v8f acc1 = {}; v8f acc2 = {}; v8f acc3 = {};

    #pragma unroll 2
    for (int k = 0; k < CCH; k += 4) {
        // A: W[M=l16][K = k + 2*hi + {0,1}]  (8-byte aligned)
        v2f a = *(const v2f*)(Arow + k + 2 * hi);

        const size_t r0 = (size_t)(k + 2 * hi) * HW;   // K row offsets in S
        const size_t r1 = r0 + (size_t)HW;

        v2f b0; b0.x = Bcol0[r0]; b0.y = Bcol0[r1];
        v2f b1; b1.x = Bcol1[r0]; b1.y = Bcol1[r1];
        v2f b2; b2.x = Bcol2[r0]; b2.y = Bcol2[r1];
        v2f b3; b3.x = Bcol3[r0]; b3.y = Bcol3[r1];

        acc0 = __builtin_amdgcn_wmma_f32_16x16x4_f32(false, a, false, b0, (short)0, acc0, false, false);
        acc1 = __builtin_amdgcn_wmma_f32_16x16x4_f32(false, a, false, b1, (short)0, acc1, false, false);
        acc2 = __builtin_amdgcn_wmma_f32_16x16x4_f32(false, a, false, b2, (short)0, acc2, false, false);
        acc3 = __builtin_amdgcn_wmma_f32_16x16x4_f32(false, a, false, b3, (short)0, acc3, false, false);
    }

    // ---- epilogue ----
    // D layout: VGPR r -> out-ch row (mtile*16 + r + 8*hi), pixel N = p0+s*16+l16

    // pixel coordinates of this lane's 4 subtile columns (shift/mask: H pow2)
    int ys[4], xs[4];
    bool cov[4];
    #pragma unroll
    for (int s = 0; s < 4; ++s) {
        const int p = p0 + s * 16 + l16;
        ys[s] = p / H;
        xs[s] = p & (H - 1);
        cov[s] = false;
    }

    // rasterize the 16 GT boxes: quantize each box once, test 4 pixels
    const float* __restrict__ bx = boxes + b * (NBOX * 4);
    constexpr float inv_s = 1.0f / (float)STRIDE;
    #pragma unroll 4
    for (int n = 0; n < NBOX; ++n) {
        int lx = (int)floorf(bx[n * 4 + 0] * inv_s); lx = lx < H - 1 ? lx : H - 1;
        int ly = (int)floorf(bx[n * 4 + 1] * inv_s); ly = ly < H - 1 ? ly : H - 1;
        int rx = (int)floorf(bx[n * 4 + 2] * inv_s); rx = rx < H - 1 ? rx : H - 1;
        int ry = (int)floorf(bx[n * 4 + 3] * inv_s); ry = ry < H - 1 ? ry : H - 1;
        const bool degen = (lx == rx) || (ly == ry);
        #pragma unroll
        for (int s = 0; s < 4; ++s) {
            const bool inside = (ys[s] >= ly) && (ys[s] < ry) && (xs[s] >= lx) && (xs[s] < rx);
            const bool point  = (ys[s] == ly) && (xs[s] == lx);
            cov[s] = cov[s] || (degen ? point : inside);
        }
    }

    float bia[8];
    #pragma unroll
    for (int r = 0; r < 8; ++r)
        bia[r] = bias[mtile * 16 + r + 8 * hi];

    float sgt = 0.0f, sbg = 0.0f;
    v8f accs[4] = {acc0, acc1, acc2, acc3};
    #pragma unroll
    for (int s = 0; s < 4; ++s) {
        const int p = p0 + s * 16 + l16;
        const float* __restrict__ Tp = Tb + (size_t)(mtile * 16 + 8 * hi) * HW + p;
        #pragma unroll
        for (int r = 0; r < 8; ++r) {
            const float t = Tp[(size_t)r * HW];
            const float d = t - (accs[s][r] + bia[r]);
            const float d2 = d * d;
            sgt += cov[s] ? d2 : 0.0f;
            sbg += cov[s] ? 0.0f : d2;
        }
    }

    // wave32 butterfly reduction
    #pragma unroll
    for (int off = 16; off >= 1; off >>= 1) {
        sgt += __shfl_xor(sgt, off, 32);
        sbg += __shfl_xor(sbg, off, 32);
    }
    if (lane == 0) {
        atomicAdd(&acc[2 * level + 0], sgt);
        atomicAdd(&acc[2 * level + 1], sbg);
    }
}

// ---------------------------------------------------------------------------
// combine the 10 accumulators into the scalar loss
// ---------------------------------------------------------------------------
__global__ void defeat_finalize(const float* __restrict__ acc, float* __restrict__ out) {
    if (threadIdx.x == 0) {
        float loss = 0.0f;
        #pragma unroll
        for (int i = 0; i < 5; ++i) {
            loss += WGT_GT * sqrtf(acc[2 * i + 0] + 1e-8f)
                  + WGT_BG * sqrtf(acc[2 * i + 1] + 1e-8f);
        }
        out[0] = loss;
    }
}

// ---------------------------------------------------------------------------
template <int H, int STRIDE>
static inline void launch_level(const float* S, const float* T, const float* Wm,
                                const float* bi, const float* boxes, float* acc,
                                int level, hipStream_t stream) {
    constexpr int ntilesN = (H * H) / 64;
    constexpr int total   = BATCH * 16 * ntilesN;  // waves
    constexpr int blocks  = (total + 7) / 8;       // 8 waves / 256-thread block
    defeat_level_kernel<H, STRIDE><<<blocks, 256, 0, stream>>>(
        S, T, Wm, bi, boxes, acc, total, level);
}

extern "C" void kernel_launch(void* const* d_in, const int* in_sizes, int n_in,
                              void* d_out, int out_size, void* d_ws, size_t ws_size,
                              hipStream_t stream) {
    float* acc = (float*)d_ws;                     // 10 floats of scratch
    const float* boxes = (const float*)d_in[20];   // gt_bboxes [8,16,4]

    defeat_init_acc<<<1, 32, 0, stream>>>(acc);

    const float* S0 = (const float*)d_in[0];  const float* T0 = (const float*)d_in[1];
    const float* W0 = (const float*)d_in[2];  const float* b0 = (const float*)d_in[3];
    const float* S1 = (const float*)d_in[4];  const float* T1 = (const float*)d_in[5];
    const float* W1 = (const float*)d_in[6];  const float* b1 = (const float*)d_in[7];
    const float* S2 = (const float*)d_in[8];  const float* T2 = (const float*)d_in[9];
    const float* W2 = (const float*)d_in[10]; const float* b2 = (const float*)d_in[11];
    const float* S3 = (const float*)d_in[12]; const float* T3 = (const float*)d_in[13];
    const float* W3 = (const float*)d_in[14]; const float* b3 = (const float*)d_in[15];
    const float* S4 = (const float*)d_in[16]; const float* T4 = (const float*)d_in[17];
    const float* W4 = (const float*)d_in[18]; const float* b4 = (const float*)d_in[19];

    launch_level<128,   8>(S0, T0, W0, b0, boxes, acc, 0, stream);
    launch_level< 64,  16>(S1, T1, W1, b1, boxes, acc, 1, stream);
    launch_level< 32,  32>(S2, T2, W2, b2, boxes, acc, 2, stream);
    launch_level< 16,  64>(S3, T3, W3, b3, boxes, acc, 3, stream);
    launch_level<  8, 128>(S4, T4, W4, b4, boxes, acc, 4, stream);

    defeat_finalize<<<1, 32, 0, stream>>>(acc, (float*)d_out);
}